// SAGEEncoder_39659728011350
// MI455X (gfx1250) — compile-verified
//
#include <hip/hip_runtime.h>

#define N_NODES 50000
#define N_EDGES 800000
#define N_GRAPHS 64
#define H 128
#define LAYERS 3
#define CLUSTERS 16
#define OUT_DIM 128
#define BN_EPS 1e-5f

typedef __attribute__((ext_vector_type(2))) float v2f;
typedef __attribute__((ext_vector_type(8))) float v8f;

// D = A(16x4 f32) * B(4x16 f32) + C(16x16 f32), full-precision WMMA.
__device__ inline v8f wmma4(v2f a, v2f b, v8f c) {
  return __builtin_amdgcn_wmma_f32_16x16x4_f32(
      false, a, false, b, (short)0, c, false, false);
}

// ---------------- embedding gather: x[n,c] = emb[deg[n], c] ----------------
__global__ void k_gather(const float* __restrict__ emb, const int* __restrict__ deg,
                         float* __restrict__ x) {
  int i = blockIdx.x * 256 + threadIdx.x;
  if (i >= N_NODES * H) return;
  int n = i >> 7, c = i & 127;
  x[i] = emb[deg[n] * H + c];
}

// ---------------- edge scatter-add: msg[dst] += x[src]; cnt[dst] += 1 -------
__global__ void k_scatter(const float* __restrict__ x, const int* __restrict__ ei,
                          float* __restrict__ msg, float* __restrict__ cnt) {
  int e = blockIdx.x * 2 + (threadIdx.x >> 7);
  if (e >= N_EDGES) return;
  int c = threadIdx.x & 127;
  int s = ei[e];
  int d = ei[N_EDGES + e];
  atomicAdd(&msg[d * H + c], x[s * H + c]);
  if (c == 0) atomicAdd(&cnt[d], 1.0f);
}

// -------- y = (msg/max(cnt,1)) @ Wl + x @ Wr + bl ; one wave = 16x128 strip --
__global__ void k_layer_gemm(const float* __restrict__ x, const float* __restrict__ msg,
                             const float* __restrict__ cnt,
                             const float* __restrict__ Wl, const float* __restrict__ Wr,
                             const float* __restrict__ bl, float* __restrict__ y) {
  int lane = threadIdx.x;
  int lm = lane & 15;
  int hi = lane >> 4;               // 0 or 1
  int row = blockIdx.x * 16 + lm;   // A-fragment row owned by this lane
  float invc = 1.0f / fmaxf(cnt[row], 1.0f);

  v8f acc[8];
  for (int t = 0; t < 8; t++) { v8f z = {}; acc[t] = z; }

  for (int k0 = 0; k0 < H; k0 += 4) {
    int ka = k0 + hi * 2;           // K pair handled by this half-wave
    v2f am = { msg[row * H + ka] * invc, msg[row * H + ka + 1] * invc };
    v2f ax = { x[row * H + ka],          x[row * H + ka + 1] };
    for (int t = 0; t < 8; t++) {
      int ncol = t * 16 + lm;
      v2f bwl = { Wl[ka * H + ncol], Wl[(ka + 1) * H + ncol] };
      v2f bwr = { Wr[ka * H + ncol], Wr[(ka + 1) * H + ncol] };
      acc[t] = wmma4(am, bwl, acc[t]);
      acc[t] = wmma4(ax, bwr, acc[t]);
    }
  }
  for (int t = 0; t < 8; t++) {
    int ncol = t * 16 + lm;
    float bias = bl[ncol];
    for (int r = 0; r < 8; r++) {
      int m = blockIdx.x * 16 + r + hi * 8;   // C/D layout: vgpr r -> M=r(+8)
      y[m * H + ncol] = acc[t][r] + bias;
    }
  }
}

// ---------------- BatchNorm stats: per-channel sum & sumsq ----------------
__global__ void k_bnstats(const float* __restrict__ y, float* __restrict__ s,
                          float* __restrict__ sq) {
  int c = threadIdx.x;  // 128 channels
  float a = 0.f, b = 0.f;
  for (int n = blockIdx.x; n < N_NODES; n += gridDim.x) {
    float v = y[n * H + c];
    a += v; b += v * v;
  }
  atomicAdd(&s[c], a);
  atomicAdd(&sq[c], b);
}

// ---------------- BN apply + ReLU -> x ----------------
__global__ void k_bnapply(const float* __restrict__ y, const float* __restrict__ s,
                          const float* __restrict__ sq, const float* __restrict__ gamma,
                          const float* __restrict__ beta, float* __restrict__ x) {
  int i = blockIdx.x * 256 + threadIdx.x;
  if (i >= N_NODES * H) return;
  int c = i & 127;
  const float invN = 1.0f / (float)N_NODES;
  float mu = s[c] * invN;
  float var = sq[c] * invN - mu * mu;
  float v = (y[i] - mu) * rsqrtf(var + BN_EPS) * gamma[c] + beta[c];
  x[i] = fmaxf(v, 0.0f);
}

// ---------------- scores = x @ attn_W + attn_b  (N-tile = 16) ----------------
__global__ void k_scores(const float* __restrict__ x, const float* __restrict__ aW,
                         const float* __restrict__ ab, float* __restrict__ sc) {
  int lane = threadIdx.x;
  int lm = lane & 15;
  int hi = lane >> 4;
  int row = blockIdx.x * 16 + lm;
  v8f acc = {};
  for (int k0 = 0; k0 < H; k0 += 4) {
    int ka = k0 + hi * 2;
    v2f a = { x[row * H + ka], x[row * H + ka + 1] };
    v2f b = { aW[ka * CLUSTERS + lm], aW[(ka + 1) * CLUSTERS + lm] };
    acc = wmma4(a, b, acc);
  }
  float bias = ab[lm];
  for (int r = 0; r < 8; r++) {
    int m = blockIdx.x * 16 + r + hi * 8;
    sc[m * CLUSTERS + lm] = acc[r] + bias;
  }
}

// ------------- order-preserving float<->uint for atomic max -------------
__device__ inline unsigned fkey(float f) {
  unsigned u = __float_as_uint(f);
  return (u & 0x80000000u) ? ~u : (u | 0x80000000u);
}
__device__ inline float funkey(unsigned u) {
  return __uint_as_float((u & 0x80000000u) ? (u ^ 0x80000000u) : ~u);
}

__global__ void k_smax(const float* __restrict__ sc, const int* __restrict__ batch,
                       unsigned* __restrict__ smax, int* __restrict__ counts) {
  int i = blockIdx.x * 256 + threadIdx.x;
  if (i >= N_NODES * CLUSTERS) return;
  int n = i >> 4, c = i & 15;
  int b = batch[n];
  atomicMax(&smax[b * CLUSTERS + c], fkey(sc[i]));
  if (c == 0) atomicAdd(&counts[b], 1);
}

__global__ void k_exp(float* __restrict__ sc, const int* __restrict__ batch,
                      const unsigned* __restrict__ smax, float* __restrict__ denom) {
  int i = blockIdx.x * 256 + threadIdx.x;
  if (i >= N_NODES * CLUSTERS) return;
  int n = i >> 4, c = i & 15;
  int b = batch[n];
  float e = expf(sc[i] - funkey(smax[b * CLUSTERS + c]));
  sc[i] = e;                       // overwrite scores with exp values
  atomicAdd(&denom[b * CLUSTERS + c], e);
}

__global__ void k_scan(const int* __restrict__ counts, int* __restrict__ starts) {
  if (threadIdx.x == 0) {
    int a = 0;
    for (int b = 0; b < N_GRAPHS; b++) { starts[b] = a; a += counts[b]; }
  }
}

// ---- clust[b] = (w_b)^T @ x_b : A = w^T (16 x K nodes), B = x (K x 16 tile) ----
__global__ void k_clust(const float* __restrict__ x, const float* __restrict__ e,
                        const float* __restrict__ denom, const int* __restrict__ starts,
                        const int* __restrict__ counts, float* __restrict__ clust) {
  int g = blockIdx.x;
  int n0 = blockIdx.y * 16;
  int lane = threadIdx.x;
  int lm = lane & 15;
  int hi = lane >> 4;
  int ks = starts[g];
  int ke = ks + counts[g];
  float invd = 1.0f / denom[g * CLUSTERS + lm];  // A row (cluster) = lm for this lane
  v8f acc = {};
  for (int k0 = ks; k0 < ke; k0 += 4) {
    int ka = k0 + hi * 2;
    v2f a = {}; v2f b = {};
    if (ka < ke) {
      a.x = e[ka * CLUSTERS + lm] * invd;
      b.x = x[ka * H + n0 + lm];
    }
    if (ka + 1 < ke) {
      a.y = e[(ka + 1) * CLUSTERS + lm] * invd;
      b.y = x[(ka + 1) * H + n0 + lm];
    }
    acc = wmma4(a, b, acc);
  }
  for (int r = 0; r < 8; r++) {
    int c = r + hi * 8;
    clust[(g * CLUSTERS + c) * H + n0 + lm] = acc[r];
  }
}

// ---------------- out = clust(1024x128) @ out_W + out_b ----------------
__global__ void k_out(const float* __restrict__ clust, const float* __restrict__ oW,
                      const float* __restrict__ ob, float* __restrict__ out) {
  int lane = threadIdx.x;
  int lm = lane & 15;
  int hi = lane >> 4;
  int m0 = blockIdx.x * 16;
  int n0 = blockIdx.y * 16;
  int row = m0 + lm;
  v8f acc = {};
  for (int k0 = 0; k0 < H; k0 += 4) {
    int ka = k0 + hi * 2;
    v2f a = { clust[row * H + ka], clust[row * H + ka + 1] };
    v2f b = { oW[ka * OUT_DIM + n0 + lm], oW[(ka + 1) * OUT_DIM + n0 + lm] };
    acc = wmma4(a, b, acc);
  }
  float bias = ob[n0 + lm];
  for (int r = 0; r < 8; r++) {
    int m = m0 + r + hi * 8;
    out[m * OUT_DIM + n0 + lm] = acc[r] + bias;
  }
}

extern "C" void kernel_launch(void* const* d_in, const int* in_sizes, int n_in,
                              void* d_out, int out_size, void* d_ws, size_t ws_size,
                              hipStream_t stream) {
  const float* emb   = (const float*)d_in[0];
  const float* Wl    = (const float*)d_in[1];
  const float* bl    = (const float*)d_in[2];
  const float* Wr    = (const float*)d_in[3];
  const float* gamma = (const float*)d_in[4];
  const float* beta  = (const float*)d_in[5];
  const float* attnW = (const float*)d_in[6];
  const float* attnb = (const float*)d_in[7];
  const float* outW  = (const float*)d_in[8];
  const float* outb  = (const float*)d_in[9];
  const int* deg     = (const int*)d_in[10];
  const int* ei      = (const int*)d_in[11];
  const int* batch   = (const int*)d_in[12];
  float* out = (float*)d_out;

  char* ws = (char*)d_ws;
  size_t off = 0;
  auto alloc = [&](size_t bytes) -> void* {
    void* p = ws + off;
    off = (off + bytes + 255) & ~(size_t)255;
    return p;
  };
  float*    x     = (float*)alloc((size_t)N_NODES * H * 4);
  float*    y     = (float*)alloc((size_t)N_NODES * H * 4);
  float*    msg   = (float*)alloc((size_t)N_NODES * H * 4);
  float*    cnt   = (float*)alloc((size_t)N_NODES * 4);
  float*    sc    = (float*)alloc((size_t)N_NODES * CLUSTERS * 4);
  float*    bns   = (float*)alloc(H * 4);
  float*    bnq   = (float*)alloc(H * 4);
  unsigned* smax  = (unsigned*)alloc(N_GRAPHS * CLUSTERS * 4);
  float*    denom = (float*)alloc(N_GRAPHS * CLUSTERS * 4);
  int*      cntsB = (int*)alloc(N_GRAPHS * 4);
  int*      starts= (int*)alloc(N_GRAPHS * 4);
  float*    clust = (float*)alloc((size_t)N_GRAPHS * CLUSTERS * H * 4);

  k_gather<<<(N_NODES * H + 255) / 256, 256, 0, stream>>>(emb, deg, x);

  for (int l = 0; l < LAYERS; l++) {
    hipMemsetAsync(msg, 0, (size_t)N_NODES * H * 4, stream);
    hipMemsetAsync(cnt, 0, (size_t)N_NODES * 4, stream);
    k_scatter<<<(N_EDGES + 1) / 2, 256, 0, stream>>>(x, ei, msg, cnt);
    k_layer_gemm<<<N_NODES / 16, 32, 0, stream>>>(x, msg, cnt,
                                                  Wl + (size_t)l * H * H,
                                                  Wr + (size_t)l * H * H,
                                                  bl + (size_t)l * H, y);
    hipMemsetAsync(bns, 0, H * 4, stream);
    hipMemsetAsync(bnq, 0, H * 4, stream);
    k_bnstats<<<512, 128, 0, stream>>>(y, bns, bnq);
    k_bnapply<<<(N_NODES * H + 255) / 256, 256, 0, stream>>>(
        y, bns, bnq, gamma + (size_t)l * H, beta + (size_t)l * H, x);
  }

  k_scores<<<N_NODES / 16, 32, 0, stream>>>(x, attnW, attnb, sc);

  hipMemsetAsync(smax, 0, N_GRAPHS * CLUSTERS * 4, stream);
  hipMemsetAsync(denom, 0, N_GRAPHS * CLUSTERS * 4, stream);
  hipMemsetAsync(cntsB, 0, N_GRAPHS * 4, stream);
  k_smax<<<(N_NODES * CLUSTERS + 255) / 256, 256, 0, stream>>>(sc, batch, smax, cntsB);
  k_exp<<<(N_NODES * CLUSTERS + 255) / 256, 256, 0, stream>>>(sc, batch, smax, denom);
  k_scan<<<1, 32, 0, stream>>>(cntsB, starts);

  dim3 cg(N_GRAPHS, H / 16);
  k_clust<<<cg, 32, 0, stream>>>(x, sc, denom, starts, cntsB, clust);

  dim3 og(N_GRAPHS * CLUSTERS / 16, OUT_DIM / 16);
  k_out<<<og, 32, 0, stream>>>(clust, outW, outb, out);
}